// SSDChunkwiseScan_68539088109640
// MI455X (gfx1250) — compile-verified
//
#include <hip/hip_runtime.h>

// ---------------------------------------------------------------------------
// Complex-gated DeltaNet chunkwise scan for MI455X (gfx1250, wave32).
//
// Shapes: B=1, L=1024, H=8, D=64, N=64 (M=32 complex), chunk C=64, nc=16.
// Output = full state at every t: 134 MB f32 -> store-bandwidth bound
// (~6 us @ 23.3 TB/s). All per-step dot products are removed via the WY
// (Householder-product) representation so the compute collapses into
// f32 WMMA GEMMs (V_WMMA_F32_16X16X4_F32) + two 64-step triangular solves,
// and the output loop is a pure streaming prefix accumulation with b128
// stores.
// ---------------------------------------------------------------------------

typedef float v2f __attribute__((ext_vector_type(2)));
typedef float v8f __attribute__((ext_vector_type(8)));

#define CC 64    // chunk length
#define HH 8     // heads
#define MM 32    // complex state rows (N/2)
#define DD 64    // key dim
#define NCHUNK 16

// workspace layout (float offsets), one block per (chunk*8 + head)
#define WS_Y_RE   0
#define WS_Y_IM   2048
#define WS_U      4096
#define WS_CA_RE  8192
#define WS_CA_IM  10240
#define WS_FH_RE  12288
#define WS_FH_IM  14336
#define WS_TA_RE  16384
#define WS_TA_IM  16416
#define WS_BLK    16448
#define WS_SIN_BASE (128 * WS_BLK)
#define WS_SIN_BLK  4096   // s_in per (c,h): [m*64+d] re, +2048 im

// One 16x16 f32 output tile via V_WMMA_F32_16X16X4_F32, accumulating over
// kdim in steps of 4.  Generic strides let us feed transposed LDS operands.
//   A element (r,k) at A[r*rsA + k*csA] ; B element (k,n) at B[k*rsB + n*csB]
// 16x4 f32 A layout (ISA 7.12.2): lanes 0-15 rows, VGPR0/1 = K {0,1} (lanes<16)
// or K {2,3} (lanes>=16).  B 4x16 mirrors it.
__device__ __forceinline__ v8f wmma_tile(const float* A, int rsA, int csA,
                                         const float* B, int rsB, int csB,
                                         int rowBase, int colBase, int kdim,
                                         int lane) {
  v8f acc = {0.f, 0.f, 0.f, 0.f, 0.f, 0.f, 0.f, 0.f};
  const int half = lane >> 4;
  const int lm   = lane & 15;
  for (int k0 = 0; k0 < kdim; k0 += 4) {
    const int ka = k0 + 2 * half;
    v2f a, b;
    a.x = A[(rowBase + lm) * rsA + ka * csA];
    a.y = A[(rowBase + lm) * rsA + (ka + 1) * csA];
    b.x = B[ka * rsB + (colBase + lm) * csB];
    b.y = B[(ka + 1) * rsB + (colBase + lm) * csB];
    acc = __builtin_amdgcn_wmma_f32_16x16x4_f32(false, a, false, b,
                                                (short)0, acc, false, false);
  }
  return acc;
}

// --------------------------------------------------------------------------
// K1: per (chunk, head).  Computes G = K K^T (WMMA), cumA (complex cumprod),
// triangular solves for y (value path) and u (correction path), and
// finalH = cumA_63 (.) (Y^T K) (WMMA).  Spills y,u,cumA,finalH,totalA to ws.
// --------------------------------------------------------------------------
extern "C" __global__ __launch_bounds__(256)
void k1_intra(const float* __restrict__ Ag, const float* __restrict__ Kg,
              const float* __restrict__ Vg, const float* __restrict__ Bg,
              float* __restrict__ ws) {
  __shared__ float lK[CC * DD];     // K[t][d]
  __shared__ float lG[CC * CC];     // G[tau][t]
  __shared__ float lAre[CC * MM];   // a -> cumA (in place)
  __shared__ float lAim[CC * MM];
  __shared__ float lYre[CC * MM];   // q -> y (in place)
  __shared__ float lYim[CC * MM];
  __shared__ float lU[CC * DD];     // u[t][d]
  __shared__ float lBeta[CC];

  const int cid = blockIdx.x;            // c*8 + h
  const int c = cid >> 3, h = cid & 7;
  const int tid = threadIdx.x, lane = tid & 31, wv = tid >> 5;
  const long gbase = ((long)c * CC * HH + h) * DD;  // elem offset of (c,t=0,h,0)
  const long trow = (long)HH * DD;                  // 512

  for (int i = tid; i < CC * DD; i += 256) {
    int t = i >> 6, d = i & 63;
    lK[i] = Kg[gbase + t * trow + d];
  }
  for (int i = tid; i < CC * MM; i += 256) {
    int t = i >> 5, m = i & 31;
    lAre[i] = Ag[gbase + t * trow + 2 * m];
    lAim[i] = Ag[gbase + t * trow + 2 * m + 1];
  }
  if (tid < CC) lBeta[tid] = Bg[(long)c * CC * HH + (long)tid * HH + h];
  __syncthreads();

  // ---- Gram matrix G = K K^T : 16 tiles, 2 per wave, all via WMMA ----
  for (int rep = 0; rep < 2; ++rep) {
    int tile = wv * 2 + rep;
    int I = tile >> 2, J = tile & 3;
    // A[i,d] = lK[i*DD+d];  B[d,j] = K[j,d] = lK[j*DD+d]  (rsB=1, csB=DD)
    v8f acc = wmma_tile(lK, DD, 1, lK, 1, DD, I * 16, J * 16, DD, lane);
    int half = lane >> 4, lm = lane & 15;
#pragma unroll
    for (int v = 0; v < 8; ++v)
      lG[(I * 16 + v + 8 * half) * CC + J * 16 + lm] = acc[v];
  }
  __syncthreads();

  // ---- cumA: complex cumprod over t (wave 0, lane = m) ----
  if (wv == 0) {
    float cr = 1.f, ci = 0.f;
    for (int t = 0; t < CC; ++t) {
      float ar = lAre[t * MM + lane], ai = lAim[t * MM + lane];
      float nr = cr * ar - ci * ai, ni = cr * ai + ci * ar;
      cr = nr; ci = ni;
      lAre[t * MM + lane] = cr;
      lAim[t * MM + lane] = ci;
    }
  }
  __syncthreads();

  // ---- q_t = b_t * v_t / cumA_t  (complex divide, clamped) ----
  for (int i = tid; i < CC * MM; i += 256) {
    int t = i >> 5, m = i & 31;
    float vr = Vg[gbase + t * trow + 2 * m];
    float vi = Vg[gbase + t * trow + 2 * m + 1];
    float cr = lAre[i], ci = lAim[i];
    float inv = 1.0f / fmaxf(cr * cr + ci * ci, 1e-30f);
    float b = lBeta[t];
    lYre[i] = b * (vr * cr + vi * ci) * inv;
    lYim[i] = b * (vi * cr - vr * ci) * inv;
  }
  __syncthreads();

  // ---- dual triangular solves (forward substitution), waves 0 and 1 ----
  if (wv == 0) {  // y-solve, lane = m:  y_t = q_t - b_t * sum_{tau<t} G[tau,t] y_tau
    for (int t = 1; t < CC; ++t) {
      float sr = 0.f, si = 0.f;
      for (int tau = 0; tau < t; ++tau) {
        float g = lG[tau * CC + t];
        sr += g * lYre[tau * MM + lane];
        si += g * lYim[tau * MM + lane];
      }
      float b = lBeta[t];
      lYre[t * MM + lane] -= b * sr;
      lYim[t * MM + lane] -= b * si;
    }
  } else if (wv == 1) {  // u-solve, lane = d & d+32: u_t = b_t (k_t - sum G u)
    float b0 = lBeta[0];
    lU[lane] = b0 * lK[lane];
    lU[lane + 32] = b0 * lK[lane + 32];
    for (int t = 1; t < CC; ++t) {
      float s0 = 0.f, s1 = 0.f;
      for (int tau = 0; tau < t; ++tau) {
        float g = lG[tau * CC + t];
        s0 += g * lU[tau * DD + lane];
        s1 += g * lU[tau * DD + lane + 32];
      }
      float b = lBeta[t];
      lU[t * DD + lane]      = b * (lK[t * DD + lane] - s0);
      lU[t * DD + lane + 32] = b * (lK[t * DD + lane + 32] - s1);
    }
  }
  __syncthreads();

  float* wsb = ws + (long)cid * WS_BLK;

  // ---- finalH = cumA_63 (.) (Y^T K) via WMMA: 8 tile-pairs, one per wave ----
  {
    int I = wv >> 2, J = wv & 3;
    // A[m,tau] = lY[tau*MM + m] (rsA=1, csA=MM); B[tau,d] = lK[tau*DD + d]
    v8f accR = wmma_tile(lYre, 1, MM, lK, DD, 1, I * 16, J * 16, CC, lane);
    v8f accI = wmma_tile(lYim, 1, MM, lK, DD, 1, I * 16, J * 16, CC, lane);
    int half = lane >> 4, lm = lane & 15;
#pragma unroll
    for (int v = 0; v < 8; ++v) {
      int m = I * 16 + v + 8 * half;
      int d = J * 16 + lm;
      float cr = lAre[(CC - 1) * MM + m], ci = lAim[(CC - 1) * MM + m];
      wsb[WS_FH_RE + m * DD + d] = cr * accR[v] - ci * accI[v];
      wsb[WS_FH_IM + m * DD + d] = cr * accI[v] + ci * accR[v];
    }
  }

  // ---- spill chunk-local data for K3 ----
  for (int i = tid; i < CC * MM; i += 256) {
    wsb[WS_Y_RE + i] = lYre[i];
    wsb[WS_Y_IM + i] = lYim[i];
    wsb[WS_CA_RE + i] = lAre[i];
    wsb[WS_CA_IM + i] = lAim[i];
  }
  for (int i = tid; i < CC * DD; i += 256) wsb[WS_U + i] = lU[i];
  if (tid < MM) {
    wsb[WS_TA_RE + tid] = lAre[(CC - 1) * MM + tid];
    wsb[WS_TA_IM + tid] = lAim[(CC - 1) * MM + tid];
  }
}

// --------------------------------------------------------------------------
// K2: inter-chunk scan (16 chunks).  One thread per (h,m,d) = 16384 threads.
//   s_in[c] = s ;  s = totalA[c] (.) s + finalH[c]
// --------------------------------------------------------------------------
extern "C" __global__ __launch_bounds__(256)
void k2_inter(float* __restrict__ ws) {
  const int gid = blockIdx.x * 256 + threadIdx.x;  // h*2048 + m*64 + d
  const int h = gid >> 11;
  const int md = gid & 2047;
  const int m = md >> 6;
  float sr = 0.f, si = 0.f;
  for (int c = 0; c < NCHUNK; ++c) {
    float* wsb  = ws + (long)(c * HH + h) * WS_BLK;
    float* sin_ = ws + WS_SIN_BASE + (long)(c * HH + h) * WS_SIN_BLK;
    sin_[md] = sr;
    sin_[2048 + md] = si;
    float tr = wsb[WS_TA_RE + m], ti = wsb[WS_TA_IM + m];
    float fr = wsb[WS_FH_RE + md], fi = wsb[WS_FH_IM + md];
    float nr = tr * sr - ti * si + fr;
    float ni = tr * si + ti * sr + fi;
    sr = nr; si = ni;
  }
}

// --------------------------------------------------------------------------
// K3: per (chunk, head).  Cmat = S U^T via WMMA, then stream 64 timesteps:
//   Y_t = cumA_t (.) ( prefix_sum(y k^T) + S - prefix_sum(c k^T) )
// writing interleaved real/imag rows with float4 (b128) stores.
// --------------------------------------------------------------------------
extern "C" __global__ __launch_bounds__(256)
void k3_out(const float* __restrict__ Kg, float* __restrict__ ws,
            float* __restrict__ out) {
  __shared__ float lK[CC * DD];
  __shared__ float lYre[CC * MM];
  __shared__ float lYim[CC * MM];
  __shared__ float lAre[CC * MM];
  __shared__ float lAim[CC * MM];
  __shared__ float lSre[MM * DD];
  __shared__ float lSim[MM * DD];
  __shared__ float lCre[CC * MM];   // Cmat[t][m]
  __shared__ float lCim[CC * MM];
  __shared__ float lU[CC * DD];

  const int cid = blockIdx.x;
  const int c = cid >> 3, h = cid & 7;
  const int tid = threadIdx.x, lane = tid & 31, wv = tid >> 5;
  const long gbase = ((long)c * CC * HH + h) * DD;
  const long trow = (long)HH * DD;
  float* wsb  = ws + (long)cid * WS_BLK;
  float* sin_ = ws + WS_SIN_BASE + (long)cid * WS_SIN_BLK;

  for (int i = tid; i < CC * DD; i += 256) {
    int t = i >> 6, d = i & 63;
    lK[i] = Kg[gbase + t * trow + d];
    lU[i] = wsb[WS_U + i];
  }
  for (int i = tid; i < CC * MM; i += 256) {
    lYre[i] = wsb[WS_Y_RE + i];
    lYim[i] = wsb[WS_Y_IM + i];
    lAre[i] = wsb[WS_CA_RE + i];
    lAim[i] = wsb[WS_CA_IM + i];
  }
  for (int i = tid; i < MM * DD; i += 256) {
    lSre[i] = sin_[i];
    lSim[i] = sin_[2048 + i];
  }
  __syncthreads();

  // ---- Cmat[t,m] = sum_d U[t,d] * S[m,d]  (re & im): 16 tile-jobs ----
  for (int rep = 0; rep < 2; ++rep) {
    int job = wv * 2 + rep;          // 0..15
    int part = job >> 3;             // 0 = re, 1 = im
    int tl = job & 7;
    int I = tl >> 1, J = tl & 1;     // t-tile 0..3, m-tile 0..1
    const float* S = part ? lSim : lSre;
    // A[t,d] = lU ; B[d,m] = S[m,d] (rsB=1, csB=DD)
    v8f acc = wmma_tile(lU, DD, 1, S, 1, DD, I * 16, J * 16, DD, lane);
    float* Cst = part ? lCim : lCre;
    int half = lane >> 4, lm = lane & 15;
#pragma unroll
    for (int v = 0; v < 8; ++v)
      Cst[(I * 16 + v + 8 * half) * MM + J * 16 + lm] = acc[v];
  }
  __syncthreads();

  // ---- streaming output: thread owns (m, 8 contiguous d) ----
  const int m = tid >> 3;
  const int db = (tid & 7) * 8;
  float Sr[8], Si[8];
  float aYr[8], aYi[8], aZr[8], aZi[8];
#pragma unroll
  for (int j = 0; j < 8; ++j) {
    Sr[j] = lSre[m * DD + db + j];
    Si[j] = lSim[m * DD + db + j];
    aYr[j] = aYi[j] = aZr[j] = aZi[j] = 0.f;
  }
  for (int t = 0; t < CC; ++t) {
    float yr = lYre[t * MM + m], yi = lYim[t * MM + m];
    float zr = lCre[t * MM + m], zi = lCim[t * MM + m];
    float ar = lAre[t * MM + m], ai = lAim[t * MM + m];
    float outr[8], outi[8];
#pragma unroll
    for (int j = 0; j < 8; ++j) {
      float k = lK[t * DD + db + j];
      aYr[j] += yr * k; aYi[j] += yi * k;
      aZr[j] += zr * k; aZi[j] += zi * k;
      float wr = aYr[j] + Sr[j] - aZr[j];
      float wi = aYi[j] + Si[j] - aZi[j];
      outr[j] = ar * wr - ai * wi;
      outi[j] = ar * wi + ai * wr;
    }
    // out[(((c*64+t)*8+h)*64 + row)*64 + d], row 2m = re, 2m+1 = im
    long ob = (gbase + t * trow) * 64 + (long)(2 * m) * 64 + db;
    *(float4*)(out + ob)      = make_float4(outr[0], outr[1], outr[2], outr[3]);
    *(float4*)(out + ob + 4)  = make_float4(outr[4], outr[5], outr[6], outr[7]);
    *(float4*)(out + ob + 64) = make_float4(outi[0], outi[1], outi[2], outi[3]);
    *(float4*)(out + ob + 68) = make_float4(outi[4], outi[5], outi[6], outi[7]);
  }
}

extern "C" void kernel_launch(void* const* d_in, const int* in_sizes, int n_in,
                              void* d_out, int out_size, void* d_ws,
                              size_t ws_size, hipStream_t stream) {
  const float* A = (const float*)d_in[0];   // A_bar (1,1024,8,64)
  const float* K = (const float*)d_in[1];   // K     (1,1024,8,64)
  const float* V = (const float*)d_in[2];   // V     (1,1024,8,64)
  const float* B = (const float*)d_in[3];   // beta  (1,1024,8)
  float* ws = (float*)d_ws;
  float* out = (float*)d_out;

  k1_intra<<<dim3(NCHUNK * HH), dim3(256), 0, stream>>>(A, K, V, B, ws);
  k2_inter<<<dim3(64), dim3(256), 0, stream>>>(ws);
  k3_out<<<dim3(NCHUNK * HH), dim3(256), 0, stream>>>(K, ws, out);
}